// GuidedAttentionLoss_59768764891553
// MI455X (gfx1250) — compile-verified
//
#include <hip/hip_runtime.h>

// Problem constants (match reference: B, N, T = 64, 512, 2048)
#define BATCH 64
#define NDIM  512
#define TDIM  2048
#define BDIM  256      // threads per block = 8 waves (wave32)
#define NBLK  2048     // blocks for the partial kernel
#define WAVES_PER_BLK (BDIM / 32)
#define PRE   4        // async pipeline depth (ring of 4 x 512B per wave)

typedef float float4v __attribute__((ext_vector_type(4)));
typedef int   v4i     __attribute__((vector_size(16)));   // matches builtin sig

// ---- CDNA5 async global->LDS path ------------------------------------------
#if __has_builtin(__builtin_amdgcn_global_load_async_to_lds_b128)
#define HAVE_ASYNC_LDS 1
// signature (probe-confirmed): (v4i AS1* src, v4i AS3* dst, Ii offset, Ii cpol)
#define ASYNC_COPY_B128(gp, lp)                                                \
  __builtin_amdgcn_global_load_async_to_lds_b128(                              \
      (__attribute__((address_space(1))) v4i*)(gp),                            \
      (__attribute__((address_space(3))) v4i*)(lp), 0, 0)
#else
#define HAVE_ASYNC_LDS 0
#endif

#if __has_builtin(__builtin_amdgcn_s_wait_asynccnt)
#define ASYNC_WAIT(n) __builtin_amdgcn_s_wait_asynccnt(n)
#else
#define ASYNC_WAIT(n) asm volatile("s_wait_asynccnt %0" ::"n"(n) : "memory")
#endif

// Hard barrier: LDS reads physically complete before any later async overwrite
// of the same ring slot can issue (async LDS writes are NOT ordered vs ds_load
// by any counter; in-order instruction issue + this wait makes reuse safe).
#define WAIT_DS0() asm volatile("s_wait_dscnt 0x0" ::: "memory")
// ----------------------------------------------------------------------------

// weight-accumulate for one 4-element lane slice of chunk cc
#define CONSUME_COMPUTE(vv, cc)                                                \
  do {                                                                         \
    const int t0 = (cc) * 128 + lane * 4;                                      \
    if (t0 + 4 <= ml) {                                                        \
      _Pragma("unroll") for (int k = 0; k < 4; ++k) {                          \
        const float d = x - (float)(t0 + k) * inv_ml;                          \
        acc = fmaf(1.0f - __expf(d * d * ninv2g2), (vv)[k], acc);              \
      }                                                                        \
    } else {                                                                   \
      _Pragma("unroll") for (int k = 0; k < 4; ++k) {                          \
        if (t0 + k < ml) {                                                     \
          const float d = x - (float)(t0 + k) * inv_ml;                        \
          acc = fmaf(1.0f - __expf(d * d * ninv2g2), (vv)[k], acc);            \
        }                                                                      \
      }                                                                        \
    }                                                                          \
  } while (0)

__global__ __launch_bounds__(BDIM) void ga_partial_kernel(
    const float* __restrict__ A, const float* __restrict__ gptr,
    const int* __restrict__ mel_lens, const int* __restrict__ seq_lens,
    float* __restrict__ partial) {
#if HAVE_ASYNC_LDS
  // per-wave ring: 4 slots x 32 lanes x 16B = 2 KB/wave, 16 KB/block
  __shared__ float4v buf[WAVES_PER_BLK][PRE][32];
#endif
  __shared__ float wsum[WAVES_PER_BLK];

  const int lane  = threadIdx.x & 31;
  const int wave  = threadIdx.x >> 5;
  const int gwave = blockIdx.x * WAVES_PER_BLK + wave;
  const int nwaves = NBLK * WAVES_PER_BLK;

  const float g = gptr[0];
  const float ninv2g2 = -1.0f / (2.0f * g * g);  // exponent multiplier

  float acc = 0.0f;

  // one wave per (b, n) row; rows with n >= seq_len contribute 0 -> skipped
  for (int row = gwave; row < BATCH * NDIM; row += nwaves) {
    const int b = row >> 9;           // / NDIM
    const int n = row & (NDIM - 1);
    const int sl = seq_lens[b];
    if (n >= sl) continue;            // fully masked row: no memory traffic
    const int ml = mel_lens[b];       // in [1024, 2048]
    const float x = (float)n / (float)sl;
    const float inv_ml = 1.0f / (float)ml;
    const float* rowp = A + ((size_t)row << 11);  // row * TDIM
    const int nchunks = (ml + 127) >> 7;          // 128 floats per wave-chunk
    // ceil(ml/128)*128 <= TDIM, so chunk loads never leave this row.

#if HAVE_ASYNC_LDS
    if (nchunks >= PRE) {             // always true for ml >= 512
#pragma unroll
      for (int p = 0; p < PRE; ++p)   // fill the ring
        ASYNC_COPY_B128(rowp + (size_t)p * 128 + lane * 4,
                        &buf[wave][p][lane]);
      int c = 0;
      for (; c + PRE < nchunks; ++c) {
        ASYNC_WAIT(PRE - 1);          // oldest (chunk c) landed in LDS
        float4v v = buf[wave][c & (PRE - 1)][lane];
        WAIT_DS0();                   // slot read complete -> safe to reuse
        ASYNC_COPY_B128(rowp + (size_t)(c + PRE) * 128 + lane * 4,
                        &buf[wave][c & (PRE - 1)][lane]);
        CONSUME_COMPUTE(v, c);
      }
      // epilogue: exactly PRE chunks in flight; descending waits 3,2,1,0
#define TAIL_STEP(wlev)                                                        \
      do {                                                                     \
        ASYNC_WAIT(wlev);                                                      \
        float4v v = buf[wave][c & (PRE - 1)][lane];                            \
        WAIT_DS0();                                                            \
        CONSUME_COMPUTE(v, c);                                                 \
        ++c;                                                                   \
      } while (0)
      TAIL_STEP(3);
      TAIL_STEP(2);
      TAIL_STEP(1);
      TAIL_STEP(0);
#undef TAIL_STEP
    } else {                          // defensive: short rows, no pipelining
      for (int c = 0; c < nchunks; ++c) {
        ASYNC_COPY_B128(rowp + (size_t)c * 128 + lane * 4,
                        &buf[wave][c & (PRE - 1)][lane]);
        ASYNC_WAIT(0);
        float4v v = buf[wave][c & (PRE - 1)][lane];
        WAIT_DS0();
        CONSUME_COMPUTE(v, c);
      }
    }
#else
    for (int c = 0; c < nchunks; ++c) {
      float4v v = __builtin_nontemporal_load(
          reinterpret_cast<const float4v*>(rowp) + (c * 32 + lane));
      CONSUME_COMPUTE(v, c);
    }
#endif
  }

  // wave32 reduction
#pragma unroll
  for (int off = 16; off > 0; off >>= 1) acc += __shfl_xor(acc, off, 32);
  if (lane == 0) wsum[wave] = acc;
  __syncthreads();
  if (threadIdx.x == 0) {
    float s = 0.0f;
#pragma unroll
    for (int w = 0; w < WAVES_PER_BLK; ++w) s += wsum[w];
    partial[blockIdx.x] = s;
  }
}

__global__ __launch_bounds__(256) void ga_reduce_kernel(
    const float* __restrict__ partial, float* __restrict__ out, float scale) {
  __shared__ float s[256];
  float a = 0.0f;
  for (int i = threadIdx.x; i < NBLK; i += 256) a += partial[i];
  s[threadIdx.x] = a;
  __syncthreads();
  for (int off = 128; off > 0; off >>= 1) {
    if ((int)threadIdx.x < off) s[threadIdx.x] += s[threadIdx.x + off];
    __syncthreads();
  }
  if (threadIdx.x == 0) out[0] = s[0] * scale;
}

extern "C" void kernel_launch(void* const* d_in, const int* in_sizes, int n_in,
                              void* d_out, int out_size, void* d_ws,
                              size_t ws_size, hipStream_t stream) {
  const float* A   = (const float*)d_in[0];
  const float* g   = (const float*)d_in[1];
  const int* mel   = (const int*)d_in[2];
  const int* seq   = (const int*)d_in[3];
  float* out       = (float*)d_out;
  float* partials  = (float*)d_ws;  // NBLK floats = 8 KB scratch

  ga_partial_kernel<<<NBLK, BDIM, 0, stream>>>(A, g, mel, seq, partials);
  ga_reduce_kernel<<<1, 256, 0, stream>>>(partials, out, 1.0f / (float)BATCH);
}